// PolarTransform_47227460387546
// MI455X (gfx1250) — compile-verified
//
#include <hip/hip_runtime.h>
#include <math.h>

// Problem constants (from reference)
#define H_IN   256
#define W_IN   256
#define NPIX   (H_IN * W_IN)      // 65536
#define NR     32                 // radii bins
#define NA     64                 // angle bins
#define NBINS  (NR * NA)          // 2048
#define PCOLS  66                 // angle dim after reflect pad (1,1)
#define PSTRIDE 80                // 66 padded to 5 * 16 (WMMA N tiles)
#define KB     68                 // 66 padded to multiple of 4 (WMMA K steps)
#define OH     128
#define OW     128

typedef __attribute__((ext_vector_type(2))) float v2f;
typedef __attribute__((ext_vector_type(8))) float v8f;

// One block per (n,c) image. 256 threads = 8 wave32.
__global__ __launch_bounds__(256) void polar_pool_resize_wmma(
    const float* __restrict__ x,        // [48, 65536]
    const float* __restrict__ counts,   // [2048]
    float* __restrict__ out)            // [48, 128, 128]
{
    __shared__ float sA[OH * NR];        // 128x32 row-interp weights (M x K)
    __shared__ float sW[KB * OW];        // 68x128 col-interp weights (K x N)
    __shared__ float sBins[NBINS];       // 2048 bin accumulators
    __shared__ float sP[NR * PSTRIDE];   // 32x80 normalized + reflect-padded
    __shared__ float sT[OH * PSTRIDE];   // 128x80 intermediate (A @ P)

    const int tid  = threadIdx.x;
    const int img  = blockIdx.x;
    const int lane = tid & 31;
    const int wave = tid >> 5;

    // ---------------- phase 0a: zero accumulated LDS regions ----------------
    for (int i = tid; i < OH * NR; i += 256) sA[i] = 0.f;
    for (int i = tid; i < KB * OW; i += 256) sW[i] = 0.f;
    for (int i = tid; i < NBINS;  i += 256) sBins[i] = 0.f;
    __syncthreads();

    // ---------------- phase 0b: build bilinear interp matrices --------------
    // Half-pixel centers (align_corners=False upsampling; matches jax triangle
    // kernel for scale > 1 including edge-clamp normalization).
    if (tid < OH) {
        // rows: 32 -> 128
        float sr = (tid + 0.5f) * ((float)NR / (float)OH) - 0.5f;
        int   i0 = (int)floorf(sr);
        float wr = sr - (float)i0;
        int r0 = min(max(i0,     0), NR - 1);
        int r1 = min(max(i0 + 1, 0), NR - 1);
        sA[tid * NR + r0] += 1.f - wr;
        sA[tid * NR + r1] += wr;
        // cols: 66 -> 128 (thread `tid` owns output column tid)
        float sc = (tid + 0.5f) * ((float)PCOLS / (float)OW) - 0.5f;
        int   c0 = (int)floorf(sc);
        float wc = sc - (float)c0;
        int cc0 = min(max(c0,     0), PCOLS - 1);
        int cc1 = min(max(c0 + 1, 0), PCOLS - 1);
        sW[cc0 * OW + tid] += 1.f - wc;
        sW[cc1 * OW + tid] += wc;
    }

    // ---------------- phase 1: analytic polar pooling (LDS atomics) ---------
    // searchsorted on uniform linspace bins == floor divide.
    const float* xi   = x + (size_t)img * NPIX;
    const float  RSCL = (float)NR / 182.0f;                // 1 / radius bin width
    const float  ASCL = (float)NA / (2.0f * (float)M_PI);  // 1 / angle bin width
    for (int p = tid; p < NPIX; p += 256) {
        int   py = p >> 8;
        int   px = p & 255;
        float yy = 128.f - (float)py;
        float xx = (float)px - 128.f;
        float d  = sqrtf(xx * xx + yy * yy);
        float a  = atan2f(yy, xx);
        if (yy < 0.f) a += 2.f * (float)M_PI;   // -> [0, 2pi)
        int ri = (int)(d * RSCL); ri = min(ri, NR - 1);
        int ai = (int)(a * ASCL); ai = min(ai, NA - 1);
        atomicAdd(&sBins[ri * NA + ai], xi[p]); // ds_add_f32
    }
    __syncthreads();

    // ------- phase 2: normalize by counts, reflect-pad angle into sP --------
    for (int i = tid; i < NR * PSTRIDE; i += 256) {
        int r = i / PSTRIDE;
        int c = i - r * PSTRIDE;
        float v = 0.f;
        if (c < PCOLS) {
            // reflect pad (1,1): col0 = orig[1], col65 = orig[62]
            int j = (c == 0) ? 1 : ((c == PCOLS - 1) ? (NA - 2) : (c - 1));
            int b = r * NA + j;
            v = sBins[b] / counts[b];
        }
        sP[i] = v;
    }
    __syncthreads();

    // WMMA f32 16x16x4 lane mapping (wave32):
    //  A frag (16x4): lanes 0-15 hold M=lane, {K=k0+0, k0+1}; lanes 16-31 hold {k0+2, k0+3}
    //  B frag (4x16): lanes 0-15 hold N=lane, rows {k0+0, k0+1}; lanes 16-31 rows {k0+2, k0+3}
    //  C/D (16x16):   vgpr v, lanes 0-15 -> M=v, lanes 16-31 -> M=v+8, N=lane&15
    const int m15  = lane & 15;
    const int kh   = (lane >> 4) << 1;  // 0 or 2
    const int mrow = (lane >> 4) << 3;  // 0 or 8

    // ---------- phase 3: stage A  T[128x80] = A[128x32] @ P[32x80] ----------
    // 8 m-tiles x 5 n-tiles = 40 tiles over 8 waves (wave-uniform, EXEC all 1s)
    for (int t = wave; t < 40; t += 8) {
        int mt = t / 5, nt = t - mt * 5;
        v8f acc = {};
        for (int k0 = 0; k0 < NR; k0 += 4) {
            v2f a, b;
            a.x = sA[(mt * 16 + m15) * NR + k0 + kh];
            a.y = sA[(mt * 16 + m15) * NR + k0 + kh + 1];
            b.x = sP[(k0 + kh)     * PSTRIDE + nt * 16 + m15];
            b.y = sP[(k0 + kh + 1) * PSTRIDE + nt * 16 + m15];
            acc = __builtin_amdgcn_wmma_f32_16x16x4_f32(
                      false, a, false, b, (short)0, acc, false, false);
        }
        #pragma unroll
        for (int v = 0; v < 8; ++v)
            sT[(mt * 16 + mrow + v) * PSTRIDE + nt * 16 + m15] = acc[v];
    }
    __syncthreads();

    // ---------- phase 4: stage B  Out[128x128] = T[128x66] @ W[66x128] ------
    // K padded to 68 (T cols 66-67 are exact zeros since P cols >=66 are zero).
    float* outp = out + (size_t)img * (OH * OW);
    for (int t = wave; t < 64; t += 8) {
        int mt = t >> 3, nt = t & 7;
        v8f acc = {};
        for (int k0 = 0; k0 < KB; k0 += 4) {
            v2f a, b;
            a.x = sT[(mt * 16 + m15) * PSTRIDE + k0 + kh];
            a.y = sT[(mt * 16 + m15) * PSTRIDE + k0 + kh + 1];
            b.x = sW[(k0 + kh)     * OW + nt * 16 + m15];
            b.y = sW[(k0 + kh + 1) * OW + nt * 16 + m15];
            acc = __builtin_amdgcn_wmma_f32_16x16x4_f32(
                      false, a, false, b, (short)0, acc, false, false);
        }
        #pragma unroll
        for (int v = 0; v < 8; ++v)
            outp[(mt * 16 + mrow + v) * OW + nt * 16 + m15] = acc[v];
    }
}

extern "C" void kernel_launch(void* const* d_in, const int* in_sizes, int n_in,
                              void* d_out, int out_size, void* d_ws, size_t ws_size,
                              hipStream_t stream) {
    const float* x      = (const float*)d_in[0];  // [16,3,256,256]
    // d_in[1] = pooling_masks [2048,65536] — intentionally unused (536 MB of
    // redundant traffic; the binary partition is reconstructed analytically)
    const float* counts = (const float*)d_in[2];  // [2048]
    float* out = (float*)d_out;                   // [16,3,128,128]

    int nimg = in_sizes[0] / NPIX;                // 48 = 16*3
    polar_pool_resize_wmma<<<nimg, 256, 0, stream>>>(x, counts, out);
}